// MPool_18090402251336
// MI455X (gfx1250) — compile-verified
//
#include <hip/hip_runtime.h>
#include <hip/hip_bf16.h>

// ---------------------------------------------------------------------------
// MPool GNN forward for MI455X (gfx1250, wave32).
// All GEMMs run on V_WMMA_F32_16X16X4_F32 (exact fp32 matrix pipe).
// Wave tile: 32x64 (8 accumulators), block: 4 waves = 128x64.
// k-loop unrolled x2 with ping-pong fragment buffers (no copy moves) and
// global_prefetch_b8 two steps ahead.
// ---------------------------------------------------------------------------

typedef __attribute__((ext_vector_type(2))) float v2f;
typedef __attribute__((ext_vector_type(8))) float v8f;

#define EPSF 1e-15f

// ------------------------------ reductions ---------------------------------
__device__ __forceinline__ float block_reduce_sum(float v, float* red) {
  int tid = threadIdx.x;
  red[tid] = v;
  __syncthreads();
  for (int s = blockDim.x >> 1; s > 0; s >>= 1) {
    if (tid < s) red[tid] += red[tid + s];
    __syncthreads();
  }
  return red[0];
}

// ------------------------------ fill ---------------------------------------
__global__ void fill_k(float* p, float v, long n) {
  long i = (long)blockIdx.x * blockDim.x + threadIdx.x;
  if (i < n) p[i] = v;
}

// --------------------------- WMMA fp32 GEMM --------------------------------
// C[b] = act( opA(A[b]) @ B[b]  (+ A2 @ B2)  (+ bias) )
// TA: logical A (M x K) is stored row-major as (K x M)  [s^T einsums].
// NG: guard N (only the final 64x10 logits GEMM; all other N % 64 == 0).
// Requirements exploited: M % 32 == 0 per wave (all M here % 64 == 0),
// K % 8 == 0, ldc == N.
template <bool TA, bool NG>
__global__ __launch_bounds__(128) void wmma_gemm_k(
    const float* __restrict__ A, const float* __restrict__ Bm,
    const float* __restrict__ A2, const float* __restrict__ B2,
    const float* __restrict__ bias, float* __restrict__ C,
    int M, int N, int K, int K2, long sA, long sB, long sC, int act) {
  const int lane = threadIdx.x & 31;
  const int wave = threadIdx.x >> 5;
  const int half = lane >> 4;  // which 16-lane half (K-pair selector)
  const int l    = lane & 15;
  const int b    = blockIdx.z;

  const float* Ab = A + (long)b * sA;
  const float* Bb = Bm + (long)b * sB;
  float* Cb = C + (long)b * sC;

  const int m0 = blockIdx.y * 128 + wave * 32;
  const int n0 = blockIdx.x * 64;
  if (m0 >= M) return;  // wave-uniform: EXEC stays all-ones for WMMA

  v8f acc[2][4];
#pragma unroll
  for (int i = 0; i < 2; ++i)
#pragma unroll
    for (int j = 0; j < 4; ++j) acc[i][j] = v8f{};

  // A fragment loader: rows [m0, m0+32), K-pair ka..ka+1 per lane half.
  // ta/ldK constant-folded at each call site.
  auto loadA = [&](const float* Ap, int ldK, int kb, bool ta, v2f& a0,
                   v2f& a1) {
    const int ka = kb + 2 * half;
    if (ta) {  // stored (K x M): element (k,m) at k*M + m
      a0.x = Ap[(long)ka * M + m0 + l];
      a0.y = Ap[(long)(ka + 1) * M + m0 + l];
      a1.x = Ap[(long)ka * M + m0 + 16 + l];
      a1.y = Ap[(long)(ka + 1) * M + m0 + 16 + l];
    } else {   // stored (M x K): aligned 8B pair -> global_load_b64
      a0 = *(const v2f*)(Ap + (long)(m0 + l) * ldK + ka);
      a1 = *(const v2f*)(Ap + (long)(m0 + 16 + l) * ldK + ka);
    }
  };
  auto loadB = [&](const float* Bp, int kb, v2f bv[4]) {
    const int ka = kb + 2 * half;
#pragma unroll
    for (int j = 0; j < 4; ++j) {
      const int col = n0 + 16 * j + l;
      if (NG) {
        bv[j].x = (col < N) ? Bp[(long)ka * N + col] : 0.0f;
        bv[j].y = (col < N) ? Bp[(long)(ka + 1) * N + col] : 0.0f;
      } else {
        bv[j].x = Bp[(long)ka * N + col];
        bv[j].y = Bp[(long)(ka + 1) * N + col];
      }
    }
  };
  auto mma8 = [&](const v2f& a0, const v2f& a1, const v2f bv[4]) {
#pragma unroll
    for (int j = 0; j < 4; ++j) {
      acc[0][j] = __builtin_amdgcn_wmma_f32_16x16x4_f32(
          false, a0, false, bv[j], (short)0, acc[0][j], false, false);
      acc[1][j] = __builtin_amdgcn_wmma_f32_16x16x4_f32(
          false, a1, false, bv[j], (short)0, acc[1][j], false, false);
    }
  };
  // One product: K % 8 == 0, ping-pong buffers, no clamp selects in the
  // address stream (uniform scalar branch on the refill only).
  auto product = [&](const float* Ap, const float* Bp, int Kp, int ldK,
                     bool ta) {
    v2f a0A, a1A, bvA[4];  // buffer A
    v2f a0B, a1B, bvB[4];  // buffer B
    loadA(Ap, ldK, 0, ta, a0A, a1A);
    loadB(Bp, 0, bvA);
    for (int kb = 0; kb < Kp; kb += 8) {
      loadA(Ap, ldK, kb + 4, ta, a0B, a1B);
      loadB(Bp, kb + 4, bvB);
      {  // speculative prefetch of the B panel two steps ahead
        int kp = (kb + 16 < Kp) ? kb + 16 : (Kp - 4);
        __builtin_prefetch(Bp + (long)(kp + 2 * half) * N + n0 + l, 0, 1);
      }
      mma8(a0A, a1A, bvA);
      if (kb + 8 < Kp) {
        loadA(Ap, ldK, kb + 8, ta, a0A, a1A);
        loadB(Bp, kb + 8, bvA);
      }
      mma8(a0B, a1B, bvB);
    }
  };

  product(Ab, Bb, K, K, TA);
  if (A2 != nullptr) product(A2, B2, K2, K2, false);

  // ---- epilogue: bias + act + store (C layout: VGPR r -> M = r + 8*half) ----
#pragma unroll
  for (int j = 0; j < 4; ++j) {
    const int col = n0 + 16 * j + l;
    if (!NG || col < N) {
      const float bb = bias ? bias[col] : 0.0f;
#pragma unroll
      for (int i = 0; i < 2; ++i) {
#pragma unroll
        for (int r = 0; r < 8; ++r) {
          const int row = m0 + 16 * i + r + 8 * half;
          float v = acc[i][j][r] + bb;
          if (act == 1) v = fmaxf(v, 0.0f);
          Cb[(long)row * N + col] = v;
        }
      }
    }
  }
}

// --------------------------- graph kernels ---------------------------------
// segment_sum of masked messages: agg[dst] += x[src] * mask
// grid = nE blocks, block = H threads
__global__ void edge_scatter_k(const float* __restrict__ x,
                               const int* __restrict__ src,
                               const int* __restrict__ dst,
                               const float* __restrict__ mask,
                               float* __restrict__ agg, int H) {
  int e = blockIdx.x;
  int f = threadIdx.x;
  float w = mask ? mask[e] : 1.0f;
  if (w == 0.0f) return;
  atomicAdd(&agg[(long)dst[e] * H + f], x[(long)src[e] * H + f] * w);
}

// dense adjacency + out-degree from edge list
__global__ void build_adj_k(const int* __restrict__ src,
                            const int* __restrict__ dst, float* __restrict__ adj,
                            float* __restrict__ deg, int N, long nE) {
  long e = (long)blockIdx.x * blockDim.x + threadIdx.x;
  if (e >= nE) return;
  int s = src[e], d = dst[e];
  int g = s / N, sl = s % N, dl = d % N;
  atomicAdd(&adj[((long)g * N + sl) * N + dl], 1.0f);
  atomicAdd(&deg[(long)g * N + sl], 1.0f);
}

// ||w||_2
__global__ void vecnorm_k(const float* __restrict__ w, float* out, int n) {
  __shared__ float red[128];
  int tid = threadIdx.x;
  float v = (tid < n) ? w[tid] * w[tid] : 0.0f;
  float s = block_reduce_sum(v, red);
  if (tid == 0) out[0] = sqrtf(s);
}

// score[node] = dot(x[node], w) / (||w|| + 1e-16)
__global__ void node_score_k(const float* __restrict__ x,
                             const float* __restrict__ w,
                             const float* __restrict__ norm,
                             float* __restrict__ score, int H) {
  __shared__ float red[128];
  long node = blockIdx.x;
  int f = threadIdx.x;
  float s = block_reduce_sum(x[node * H + f] * w[f], red);
  if (f == 0) score[node] = s / (norm[0] + 1e-16f);
}

// Exact top-k by ranking (stable, tie-break toward lower index like lax.top_k).
__global__ void topk_rank_k(const float* __restrict__ score, int n, int k,
                            int* __restrict__ perm, int* __restrict__ new_id) {
  extern __shared__ float ssc[];
  int b = blockIdx.x, i = threadIdx.x;
  ssc[i] = score[(long)b * n + i];
  __syncthreads();
  float si = ssc[i];
  int cnt = 0;
  for (int j = 0; j < n; ++j) {
    float sj = ssc[j];
    cnt += (sj > si) || (sj == si && j < i);
  }
  if (cnt < k) {
    perm[b * k + cnt] = b * n + i;
    new_id[(long)b * n + i] = b * k + cnt;
  } else {
    new_id[(long)b * n + i] = -1;
  }
}

// new_x[r] = x[perm[r]] * tanh(score[perm[r]])
__global__ void gate_gather_k(const float* __restrict__ xin,
                              const float* __restrict__ score,
                              const int* __restrict__ perm,
                              float* __restrict__ xout, long total, int H) {
  long gid = (long)blockIdx.x * blockDim.x + threadIdx.x;
  if (gid >= total) return;
  int r = (int)(gid / H), f = (int)(gid % H);
  int p = perm[r];
  xout[gid] = xin[(long)p * H + f] * tanhf(score[p]);
}

__global__ void edge_remap_k(const int* __restrict__ new_id,
                             const int* __restrict__ srcIn,
                             const int* __restrict__ dstIn,
                             const float* __restrict__ mIn,
                             int* __restrict__ srcOut, int* __restrict__ dstOut,
                             float* __restrict__ mOut, long nE) {
  long e = (long)blockIdx.x * blockDim.x + threadIdx.x;
  if (e >= nE) return;
  int ns = new_id[srcIn[e]];
  int nd = new_id[dstIn[e]];
  float mv = mIn ? mIn[e] : 1.0f;
  bool valid = (ns >= 0) && (nd >= 0) && (mv > 0.0f);
  srcOut[e] = valid ? ns : 0;
  dstOut[e] = valid ? nd : 0;
  mOut[e] = valid ? 1.0f : 0.0f;
}

// gmp || gap readout over k contiguous nodes per graph
__global__ void readout_k(const float* __restrict__ x, int k, int H,
                          float* __restrict__ out) {
  int b = blockIdx.x, f = threadIdx.x;
  float mx = -3.402823466e38f, sm = 0.0f;
  for (int r = 0; r < k; ++r) {
    float v = x[((long)b * k + r) * H + f];
    mx = fmaxf(mx, v);
    sm += v;
  }
  out[(long)b * 2 * H + f] = mx;
  out[(long)b * 2 * H + H + f] = sm / (float)k;
}

// --------------------------- dense-branch kernels --------------------------
__global__ void softmax_rows_k(float* __restrict__ x, int cols) {
  __shared__ float red[256];
  long row = blockIdx.x;
  int tid = threadIdx.x;
  float v = x[row * cols + tid];
  red[tid] = v;
  __syncthreads();
  for (int s = blockDim.x >> 1; s > 0; s >>= 1) {
    if (tid < s) red[tid] = fmaxf(red[tid], red[tid + s]);
    __syncthreads();
  }
  float mx = red[0];
  __syncthreads();
  float e = __expf(v - mx);
  red[tid] = e;
  __syncthreads();
  for (int s = blockDim.x >> 1; s > 0; s >>= 1) {
    if (tid < s) red[tid] += red[tid + s];
    __syncthreads();
  }
  x[row * cols + tid] = e / red[0];
}

__global__ void trace_k(const float* __restrict__ m, int n, float* out) {
  __shared__ float red[256];
  int b = blockIdx.x, tid = threadIdx.x;
  float acc = 0.0f;
  for (int i = tid; i < n; i += blockDim.x) acc += m[((long)b * n + i) * n + i];
  float s = block_reduce_sum(acc, red);
  if (tid == 0) out[b] = s;
}

// den[b] = sum_i deg[b,i] * sum_k s[b,i,k]^2
__global__ void mincut_den_k(const float* __restrict__ deg,
                             const float* __restrict__ s, int n, int k,
                             float* out) {
  __shared__ float red[256];
  int b = blockIdx.x, tid = threadIdx.x;
  float acc = 0.0f;
  for (int i = tid; i < n; i += blockDim.x) {
    const float* sp = s + ((long)b * n + i) * k;
    float ss = 0.0f;
    for (int j = 0; j < k; ++j) ss += sp[j] * sp[j];
    acc += deg[(long)b * n + i] * ss;
  }
  float r = block_reduce_sum(acc, red);
  if (tid == 0) out[b] = r;
}

__global__ void frob_k(const float* __restrict__ m, int n, float* out) {
  __shared__ float red[256];
  int b = blockIdx.x, tid = threadIdx.x;
  float acc = 0.0f;
  for (long idx = tid; idx < (long)n * n; idx += blockDim.x) {
    float v = m[(long)b * n * n + idx];
    acc += v * v;
  }
  float s = block_reduce_sum(acc, red);
  if (tid == 0) out[b] = sqrtf(s);
}

// o[b] = || ss/(||ss||+EPS) - I/sqrt(k) ||_F
__global__ void ortho_k(const float* __restrict__ ss,
                        const float* __restrict__ ssn, int k, float* out) {
  __shared__ float red[256];
  int b = blockIdx.x, tid = threadIdx.x;
  float nrm = ssn[b] + EPSF;
  float diag = 1.0f / sqrtf((float)k);
  float acc = 0.0f;
  for (long idx = tid; idx < (long)k * k; idx += blockDim.x) {
    int i = (int)(idx / k), j = (int)(idx % k);
    float d = ss[(long)b * k * k + idx] / nrm - ((i == j) ? diag : 0.0f);
    acc += d * d;
  }
  float s = block_reduce_sum(acc, red);
  if (tid == 0) out[b] = sqrtf(s);
}

// zero diagonal, dpool = sqrt(rowsum) + EPS.   grid = B*n blocks of n threads
__global__ void zero_diag_rowsum_k(float* __restrict__ m, int n,
                                   float* __restrict__ dpool) {
  __shared__ float red[256];
  int r = blockIdx.x;
  int b = r / n, i = r % n, j = threadIdx.x;
  long base = ((long)b * n + i) * n;
  float v = (j == i) ? 0.0f : m[base + j];
  if (j == i) m[base + j] = 0.0f;
  float s = block_reduce_sum(v, red);
  if (j == 0) dpool[r] = sqrtf(s) + EPSF;
}

__global__ void adj_normalize_k(float* __restrict__ m,
                                const float* __restrict__ dpool, int n,
                                long total) {
  long gid = (long)blockIdx.x * blockDim.x + threadIdx.x;
  if (gid >= total) return;
  long bi = gid / n;  // b*n + i
  int j = (int)(gid % n);
  long b = bi / n;
  m[gid] /= (dpool[bi] * dpool[b * n + j]);
}

__global__ void rowsum_k(const float* __restrict__ m, int n,
                         float* __restrict__ out) {
  __shared__ float red[256];
  int r = blockIdx.x, j = threadIdx.x;
  float s = block_reduce_sum(m[(long)r * n + j], red);
  if (j == 0) out[r] = s;
}

// T[row,:] /= max(rs[row], 1)
__global__ void row_divide_k(float* __restrict__ T,
                             const float* __restrict__ rs, int H, long total) {
  long gid = (long)blockIdx.x * blockDim.x + threadIdx.x;
  if (gid >= total) return;
  T[gid] /= fmaxf(rs[gid / H], 1.0f);
}

__global__ void mean_rows_k(const float* __restrict__ x, int n, int H,
                            float* __restrict__ out) {
  int b = blockIdx.x, f = threadIdx.x;
  float s = 0.0f;
  for (int i = 0; i < n; ++i) s += x[((long)b * n + i) * H + f];
  out[(long)b * H + f] = s / (float)n;
}

__global__ void add3_k(const float* a, const float* b, const float* c,
                       float* out, long n) {
  long i = (long)blockIdx.x * blockDim.x + threadIdx.x;
  if (i < n) out[i] = a[i] + b[i] + c[i];
}

__global__ void concat_k(const float* a, const float* b, float* out, int H,
                         long total) {
  long gid = (long)blockIdx.x * blockDim.x + threadIdx.x;
  if (gid >= total) return;
  long row = gid / (2 * H);
  int f = (int)(gid % (2 * H));
  out[gid] = (f < H) ? a[row * H + f] : b[row * H + (f - H)];
}

__global__ void finalize_k(const float* num1, const float* den1, const float* o1,
                           const float* num2, const float* den2, const float* o2,
                           float* out) {
  __shared__ float r1[64], r2[64];
  int b = threadIdx.x;  // 64 threads
  r1[b] = num1[b] / (den1[b] + EPSF) + num2[b] / (den2[b] + EPSF);
  r2[b] = o1[b] + o2[b];
  __syncthreads();
  for (int s = 32; s > 0; s >>= 1) {
    if (b < s) { r1[b] += r1[b + s]; r2[b] += r2[b + s]; }
    __syncthreads();
  }
  if (b == 0) {
    out[640] = -(r1[0] / 64.0f);  // mc = mc1 + mc2
    out[641] = r2[0] / 64.0f;     // o  = o1 + o2
  }
}

// ---------------------------------------------------------------------------
extern "C" void kernel_launch(void* const* d_in, const int* in_sizes, int n_in,
                              void* d_out, int out_size, void* d_ws,
                              size_t ws_size, hipStream_t stream) {
  (void)in_sizes; (void)n_in; (void)out_size; (void)ws_size;
  constexpr int B = 64, N = 512, E = 8192, H = 128, K1 = 256, K2 = 128;
  constexpr long BN = (long)B * N;  // 32768
  constexpr long BE = (long)B * E;  // 524288

  const float* x         = (const float*)d_in[0];
  const int*   src       = (const int*)d_in[1];
  const int*   dstp      = (const int*)d_in[2];
  const float* conv0_wr  = (const float*)d_in[3];
  const float* conv0_wrt = (const float*)d_in[4];
  const float* conv0_b   = (const float*)d_in[5];
  const float* sc2_wr    = (const float*)d_in[6];
  const float* sc2_wrt   = (const float*)d_in[7];
  const float* sc2_b     = (const float*)d_in[8];
  const float* sc3_wr    = (const float*)d_in[9];
  const float* sc3_wrt   = (const float*)d_in[10];
  const float* sc3_b     = (const float*)d_in[11];
  const float* sg1_wr    = (const float*)d_in[12];
  const float* sg1_wrt   = (const float*)d_in[13];
  const float* sg1_b     = (const float*)d_in[14];
  const float* sg2_wr    = (const float*)d_in[15];
  const float* sg2_wrt   = (const float*)d_in[16];
  const float* sg2_b     = (const float*)d_in[17];
  const float* pool0_w   = (const float*)d_in[18];
  const float* pool0_b   = (const float*)d_in[19];
  const float* pool1_w   = (const float*)d_in[20];
  const float* pool1_b   = (const float*)d_in[21];
  const float* p1        = (const float*)d_in[22];
  const float* p2        = (const float*)d_in[23];
  const float* p3        = (const float*)d_in[24];
  const float* lin11_w   = (const float*)d_in[25];
  const float* lin11_b   = (const float*)d_in[26];
  const float* lin1_w    = (const float*)d_in[27];
  const float* lin1_b    = (const float*)d_in[28];
  const float* lin2_w    = (const float*)d_in[29];
  const float* lin2_b    = (const float*)d_in[30];
  float* out = (float*)d_out;

  // ---- bump allocator on d_ws ----
  size_t off = 0;
  auto allocF = [&](size_t n) -> float* {
    float* p = (float*)((char*)d_ws + off);
    off += ((n * 4 + 255) / 256) * 256;
    return p;
  };
  auto allocI = [&](size_t n) -> int* {
    int* p = (int*)((char*)d_ws + off);
    off += ((n * 4 + 255) / 256) * 256;
    return p;
  };

  float* norms  = allocF(4);
  float* agg0   = allocF(BN * H);
  float* x0     = allocF(BN * H);
  float* score1 = allocF(BN);
  float* xA     = allocF((long)B * K1 * H);  // stage-1 pooled x
  float* sx1    = allocF((long)B * 2 * H);
  int*   perm1  = allocI((long)B * K1);
  int*   newid1 = allocI(BN);
  int*   s1     = allocI(BE);
  int*   d1     = allocI(BE);
  float* m1     = allocF(BE);
  float* adj    = allocF((long)B * N * N);
  float* deg    = allocF(BN);
  float* sbuf   = allocF(BN * K1);  // s (softmax in place)
  float* T1     = allocF(BN * K1);  // adj@s, later reused for ss1
  float* out1   = allocF((long)B * K1 * H);
  float* oadj1  = allocF((long)B * K1 * K1);
  float* num1   = allocF(64); float* den1 = allocF(64);
  float* ssn1   = allocF(64); float* o1v  = allocF(64);
  float* dpool1 = allocF((long)B * K1);
  float* deg2   = allocF((long)B * K1);
  float* Tsg1   = allocF((long)B * K1 * H);
  float* xd1    = allocF((long)B * K1 * H);
  float* s2buf  = allocF((long)B * K1 * K2);
  float* out2   = allocF((long)B * K2 * H);
  float* T2     = allocF((long)B * K1 * K2);  // adj1@s2, later reused for ss2
  float* oadj2  = allocF((long)B * K2 * K2);
  float* num2   = allocF(64); float* den2 = allocF(64);
  float* ssn2   = allocF(64); float* o2v  = allocF(64);
  float* dpool2 = allocF((long)B * K2);
  float* deg3   = allocF((long)B * K2);
  float* Tsg2   = allocF((long)B * K2 * H);
  float* xd2    = allocF((long)B * K2 * H);
  float* xmean  = allocF((long)B * H);
  float* xg     = allocF((long)B * H);
  float* aggS   = allocF((long)B * K1 * H);
  float* cv2    = allocF((long)B * K1 * H);
  float* score2 = allocF((long)B * K1);
  int*   perm2  = allocI((long)B * K2);
  int*   newid2 = allocI((long)B * K1);
  int*   s2e    = allocI(BE);
  int*   d2e    = allocI(BE);
  float* m2e    = allocF(BE);
  float* xB     = allocF((long)B * K2 * H);
  float* sx2    = allocF((long)B * 2 * H);
  float* aggT   = allocF((long)B * K2 * H);
  float* cv3    = allocF((long)B * K2 * H);
  float* score3 = allocF((long)B * K2);
  int*   perm3  = allocI((long)B * 64);
  int*   newid3 = allocI((long)B * K2);
  float* xC     = allocF((long)B * 64 * H);
  float* sx3    = allocF((long)B * 2 * H);
  float* sxsum  = allocF((long)B * 2 * H);
  float* sxr    = allocF((long)B * H);
  float* cat    = allocF((long)B * 2 * H);

  auto nblk = [](long n, int t) { return (unsigned)((n + t - 1) / t); };
  auto zero = [&](float* p, long n) {
    fill_k<<<nblk(n, 256), 256, 0, stream>>>(p, 0.0f, n);
  };
  auto gemm = [&](bool TA, const float* A, const float* Bw, const float* A2,
                  const float* B2, int K2v, const float* bias, float* C, int M,
                  int Nn, int K, int batch, long sA, long sB, long sC,
                  int act) {
    dim3 g((unsigned)((Nn + 63) / 64), (unsigned)((M + 127) / 128),
           (unsigned)batch);
    bool ng = (Nn % 64) != 0;
    if (TA)
      wmma_gemm_k<true, false><<<g, 128, 0, stream>>>(A, Bw, A2, B2, bias, C, M,
                                                      Nn, K, K2v, sA, sB, sC,
                                                      act);
    else if (ng)
      wmma_gemm_k<false, true><<<g, 128, 0, stream>>>(A, Bw, A2, B2, bias, C, M,
                                                      Nn, K, K2v, sA, sB, sC,
                                                      act);
    else
      wmma_gemm_k<false, false><<<g, 128, 0, stream>>>(A, Bw, A2, B2, bias, C,
                                                       M, Nn, K, K2v, sA, sB,
                                                       sC, act);
  };

  // ---- score-vector norms ----
  vecnorm_k<<<1, 128, 0, stream>>>(p1, norms + 0, H);
  vecnorm_k<<<1, 128, 0, stream>>>(p2, norms + 1, H);
  vecnorm_k<<<1, 128, 0, stream>>>(p3, norms + 2, H);

  // ---- conv0: x0 = relu(seg_sum(x[src])@wr + x@wroot + b) ----
  zero(agg0, BN * H);
  edge_scatter_k<<<(unsigned)BE, H, 0, stream>>>(x, src, dstp, nullptr, agg0, H);
  gemm(false, agg0, conv0_wr, x, conv0_wrt, H, conv0_b, x0, (int)BN, H, H, 1, 0,
       0, 0, 1);

  // ---- dense adjacency + degree ----
  zero(adj, (long)B * N * N);
  zero(deg, BN);
  build_adj_k<<<nblk(BE, 256), 256, 0, stream>>>(src, dstp, adj, deg, N, BE);

  // ---- sparse top-k stage 1 (N=512 -> 256) ----
  node_score_k<<<(unsigned)BN, 128, 0, stream>>>(x0, p1, norms + 0, score1, H);
  topk_rank_k<<<B, N, N * sizeof(float), stream>>>(score1, N, K1, perm1,
                                                   newid1);
  gate_gather_k<<<nblk((long)B * K1 * H, 256), 256, 0, stream>>>(
      x0, score1, perm1, xA, (long)B * K1 * H, H);
  edge_remap_k<<<nblk(BE, 256), 256, 0, stream>>>(newid1, src, dstp, nullptr,
                                                  s1, d1, m1, BE);
  readout_k<<<B, H, 0, stream>>>(xA, K1, H, sx1);

  // ---- dense mincut pool 1 ----
  gemm(false, x0, pool0_w, nullptr, nullptr, 0, pool0_b, sbuf, (int)BN, K1, H,
       1, 0, 0, 0, 0);
  softmax_rows_k<<<(unsigned)BN, K1, 0, stream>>>(sbuf, K1);
  // out1 = s^T x   (batched, TA)
  gemm(true, sbuf, x0, nullptr, nullptr, 0, nullptr, out1, K1, H, N, B,
       (long)N * K1, (long)N * H, (long)K1 * H, 0);
  // T1 = adj @ s
  gemm(false, adj, sbuf, nullptr, nullptr, 0, nullptr, T1, N, K1, N, B,
       (long)N * N, (long)N * K1, (long)N * K1, 0);
  // oadj1 = s^T T1
  gemm(true, sbuf, T1, nullptr, nullptr, 0, nullptr, oadj1, K1, K1, N, B,
       (long)N * K1, (long)N * K1, (long)K1 * K1, 0);
  trace_k<<<B, 256, 0, stream>>>(oadj1, K1, num1);
  mincut_den_k<<<B, 256, 0, stream>>>(deg, sbuf, N, K1, den1);
  // ss1 = s^T s  (reuse T1 storage)
  gemm(true, sbuf, sbuf, nullptr, nullptr, 0, nullptr, T1, K1, K1, N, B,
       (long)N * K1, (long)N * K1, (long)K1 * K1, 0);
  frob_k<<<B, 256, 0, stream>>>(T1, K1, ssn1);
  ortho_k<<<B, 256, 0, stream>>>(T1, ssn1, K1, o1v);
  zero_diag_rowsum_k<<<(unsigned)(B * K1), K1, 0, stream>>>(oadj1, K1, dpool1);
  adj_normalize_k<<<nblk((long)B * K1 * K1, 256), 256, 0, stream>>>(
      oadj1, dpool1, K1, (long)B * K1 * K1);
  rowsum_k<<<(unsigned)(B * K1), K1, 0, stream>>>(oadj1, K1, deg2);

  // ---- dense SAGE 1 ----
  gemm(false, oadj1, out1, nullptr, nullptr, 0, nullptr, Tsg1, K1, H, K1, B,
       (long)K1 * K1, (long)K1 * H, (long)K1 * H, 0);
  row_divide_k<<<nblk((long)B * K1 * H, 256), 256, 0, stream>>>(
      Tsg1, deg2, H, (long)B * K1 * H);
  gemm(false, Tsg1, sg1_wr, out1, sg1_wrt, H, sg1_b, xd1, B * K1, H, H, 1, 0, 0,
       0, 1);

  // ---- dense mincut pool 2 ----
  gemm(false, xd1, pool1_w, nullptr, nullptr, 0, pool1_b, s2buf, B * K1, K2, H,
       1, 0, 0, 0, 0);
  softmax_rows_k<<<(unsigned)(B * K1), K2, 0, stream>>>(s2buf, K2);
  gemm(true, s2buf, xd1, nullptr, nullptr, 0, nullptr, out2, K2, H, K1, B,
       (long)K1 * K2, (long)K1 * H, (long)K2 * H, 0);
  gemm(false, oadj1, s2buf, nullptr, nullptr, 0, nullptr, T2, K1, K2, K1, B,
       (long)K1 * K1, (long)K1 * K2, (long)K1 * K2, 0);
  gemm(true, s2buf, T2, nullptr, nullptr, 0, nullptr, oadj2, K2, K2, K1, B,
       (long)K1 * K2, (long)K1 * K2, (long)K2 * K2, 0);
  trace_k<<<B, 256, 0, stream>>>(oadj2, K2, num2);
  mincut_den_k<<<B, 256, 0, stream>>>(deg2, s2buf, K1, K2, den2);
  gemm(true, s2buf, s2buf, nullptr, nullptr, 0, nullptr, T2, K2, K2, K1, B,
       (long)K1 * K2, (long)K1 * K2, (long)K2 * K2, 0);  // ss2 into T2
  frob_k<<<B, 256, 0, stream>>>(T2, K2, ssn2);
  ortho_k<<<B, 256, 0, stream>>>(T2, ssn2, K2, o2v);
  zero_diag_rowsum_k<<<(unsigned)(B * K2), K2, 0, stream>>>(oadj2, K2, dpool2);
  adj_normalize_k<<<nblk((long)B * K2 * K2, 256), 256, 0, stream>>>(
      oadj2, dpool2, K2, (long)B * K2 * K2);
  rowsum_k<<<(unsigned)(B * K2), K2, 0, stream>>>(oadj2, K2, deg3);

  // ---- dense SAGE 2 (no relu) + graph readout ----
  gemm(false, oadj2, out2, nullptr, nullptr, 0, nullptr, Tsg2, K2, H, K2, B,
       (long)K2 * K2, (long)K2 * H, (long)K2 * H, 0);
  row_divide_k<<<nblk((long)B * K2 * H, 256), 256, 0, stream>>>(
      Tsg2, deg3, H, (long)B * K2 * H);
  gemm(false, Tsg2, sg2_wr, out2, sg2_wrt, H, sg2_b, xd2, B * K2, H, H, 1, 0, 0,
       0, 0);
  mean_rows_k<<<B, H, 0, stream>>>(xd2, K2, H, xmean);
  gemm(false, xmean, lin1_w, nullptr, nullptr, 0, lin1_b, xg, B, H, H, 1, 0, 0,
       0, 1);

  // ---- sparse stage 2 (256 -> 128) ----
  zero(aggS, (long)B * K1 * H);
  edge_scatter_k<<<(unsigned)BE, H, 0, stream>>>(xA, s1, d1, m1, aggS, H);
  gemm(false, aggS, sc2_wr, xA, sc2_wrt, H, sc2_b, cv2, B * K1, H, H, 1, 0, 0,
       0, 1);
  node_score_k<<<(unsigned)(B * K1), 128, 0, stream>>>(cv2, p2, norms + 1,
                                                       score2, H);
  topk_rank_k<<<B, K1, K1 * sizeof(float), stream>>>(score2, K1, K2, perm2,
                                                     newid2);
  gate_gather_k<<<nblk((long)B * K2 * H, 256), 256, 0, stream>>>(
      cv2, score2, perm2, xB, (long)B * K2 * H, H);
  edge_remap_k<<<nblk(BE, 256), 256, 0, stream>>>(newid2, s1, d1, m1, s2e, d2e,
                                                  m2e, BE);
  readout_k<<<B, H, 0, stream>>>(xB, K2, H, sx2);

  // ---- sparse stage 3 (128 -> 64) ----
  zero(aggT, (long)B * K2 * H);
  edge_scatter_k<<<(unsigned)BE, H, 0, stream>>>(xB, s2e, d2e, m2e, aggT, H);
  gemm(false, aggT, sc3_wr, xB, sc3_wrt, H, sc3_b, cv3, B * K2, H, H, 1, 0, 0,
       0, 1);
  node_score_k<<<(unsigned)(B * K2), 128, 0, stream>>>(cv3, p3, norms + 2,
                                                       score3, H);
  topk_rank_k<<<B, K2, K2 * sizeof(float), stream>>>(score3, K2, 64, perm3,
                                                     newid3);
  gate_gather_k<<<nblk((long)B * 64 * H, 256), 256, 0, stream>>>(
      cv3, score3, perm3, xC, (long)B * 64 * H, H);
  readout_k<<<B, H, 0, stream>>>(xC, 64, H, sx3);

  // ---- heads ----
  add3_k<<<nblk((long)B * 2 * H, 256), 256, 0, stream>>>(sx1, sx2, sx3, sxsum,
                                                         (long)B * 2 * H);
  gemm(false, sxsum, lin11_w, nullptr, nullptr, 0, lin11_b, sxr, B, H, 2 * H, 1,
       0, 0, 0, 1);
  concat_k<<<nblk((long)B * 2 * H, 256), 256, 0, stream>>>(sxr, xg, cat, H,
                                                           (long)B * 2 * H);
  gemm(false, cat, lin2_w, nullptr, nullptr, 0, lin2_b, out, B, 10, 2 * H, 1, 0,
       0, 0, 0);  // logits -> d_out[0:640]
  finalize_k<<<1, 64, 0, stream>>>(num1, den1, o1v, num2, den2, o2v, out);
}